// MLPDecoder_25220047962356
// MI455X (gfx1250) — compile-verified
//
#include <hip/hip_runtime.h>
#include <hip/hip_bf16.h>

typedef float v2f __attribute__((ext_vector_type(2)));
typedef float v8f __attribute__((ext_vector_type(8)));

#define DCONST 128
#define HIDC 64

// ---------------------------------------------------------------------------
// Pack combined weights into workspace:
//  Wn   [128 x 128] : W1 rows 384..511, cols 0..63 scorer / 64..127 type
//  Wg   [384 x 128] : W1 rows 0..383
//  Wtail[  4 x 128] : W1 rows 512..515 (dist row + 3 feature rows)
//  bias [128]       : b1 scorer ++ b1 type
//  stop_tail [64]   : b1s + no_more_edges_rep(132) . W1s[384:516]
// ---------------------------------------------------------------------------
__global__ void pack_weights_kernel(const float* __restrict__ sW1,
                                    const float* __restrict__ sb1,
                                    const float* __restrict__ tW1,
                                    const float* __restrict__ tb1,
                                    const float* __restrict__ no_more,
                                    float* __restrict__ Wn,
                                    float* __restrict__ Wg,
                                    float* __restrict__ Wtail,
                                    float* __restrict__ bias,
                                    float* __restrict__ stop_tail) {
    int i = blockIdx.x * blockDim.x + threadIdx.x;
    if (i < 128 * 128) {
        int k = i >> 7, j = i & 127;
        Wn[i] = (j < 64) ? sW1[(384 + k) * 64 + j] : tW1[(384 + k) * 64 + (j - 64)];
    }
    int i2 = i - 128 * 128;
    if (i2 >= 0 && i2 < 384 * 128) {
        int k = i2 >> 7, j = i2 & 127;
        Wg[i2] = (j < 64) ? sW1[k * 64 + j] : tW1[k * 64 + (j - 64)];
    }
    int i3 = i2 - 384 * 128;
    if (i3 >= 0 && i3 < 4 * 128) {
        int k = i3 >> 7, j = i3 & 127;
        Wtail[i3] = (j < 64) ? sW1[(512 + k) * 64 + j] : tW1[(512 + k) * 64 + (j - 64)];
    }
    int i4 = i3 - 4 * 128;
    if (i4 >= 0 && i4 < 128) {
        bias[i4] = (i4 < 64) ? sb1[i4] : tb1[i4 - 64];
    }
    int i5 = i4 - 128;
    if (i5 >= 0 && i5 < 64) {
        float s = sb1[i5];
        for (int r = 0; r < 132; ++r)
            s += no_more[r] * sW1[(384 + r) * 64 + i5];
        stop_tail[i5] = s;
    }
}

// ---------------------------------------------------------------------------
// nodeC[N,128] = node_reps[N,128] @ Wn[128,128]   via v_wmma_f32_16x16x4_f32
// One block = 256 threads = 8 waves; block owns a 16-row strip, wave w owns
// the 16-col tile w. K swept in steps of 4 (32 WMMA ops per tile).
// ---------------------------------------------------------------------------
__global__ void gemm_node_kernel(const float* __restrict__ A,   // [N,128]
                                 const float* __restrict__ B,   // [128,128]
                                 float* __restrict__ C) {       // [N,128]
    const int lane = threadIdx.x & 31;
    const int wave = threadIdx.x >> 5;
    const int ti = blockIdx.x;          // row tile
    const int tj = wave;                // col tile 0..7
    const int m = lane & 15;
    const int half = lane >> 4;         // 0 or 1

    const float* arow = A + (ti * 16 + m) * 128;
    const float* bcol = B + tj * 16 + m;
    v8f acc = {};
    for (int kb = 0; kb < 128; kb += 4) {
        v2f a, b;
        a[0] = arow[kb + 2 * half];
        a[1] = arow[kb + 2 * half + 1];
        b[0] = bcol[(kb + 2 * half) * 128];
        b[1] = bcol[(kb + 2 * half + 1) * 128];
        acc = __builtin_amdgcn_wmma_f32_16x16x4_f32(false, a, false, b,
                                                    (short)0, acc, false, false);
    }
    float* crow = C + tj * 16 + m;
#pragma unroll
    for (int r = 0; r < 8; ++r)
        crow[(ti * 16 + r + 8 * half) * 128] = acc[r];
}

// ---------------------------------------------------------------------------
// graphC[G,128] = [mol | partial | node_reps[focus]] (G x 384) @ Wg[384,128]
// ---------------------------------------------------------------------------
__global__ void gemm_graph_kernel(const float* __restrict__ mol,
                                  const float* __restrict__ par,
                                  const float* __restrict__ nodes,
                                  const int* __restrict__ focus,
                                  const float* __restrict__ B,   // [384,128]
                                  float* __restrict__ C) {       // [G,128]
    const int lane = threadIdx.x & 31;
    const int wave = threadIdx.x >> 5;
    const int tile = blockIdx.x * 8 + wave;
    const int ti = tile >> 3;           // row tile (G/16)
    const int tj = tile & 7;            // col tile
    const int m = lane & 15;
    const int half = lane >> 4;

    const int row = ti * 16 + m;
    const float* seg0 = mol + row * 128;
    const float* seg1 = par + row * 128;
    const float* seg2 = nodes + (long)focus[row] * 128;
    const float* bcol = B + tj * 16 + m;

    v8f acc = {};
    for (int kb = 0; kb < 384; kb += 4) {
        const float* seg = (kb < 128) ? seg0 : (kb < 256) ? seg1 : seg2;
        const int ko = kb & 127;
        v2f a, b;
        a[0] = seg[ko + 2 * half];
        a[1] = seg[ko + 2 * half + 1];
        b[0] = bcol[(kb + 2 * half) * 128];
        b[1] = bcol[(kb + 2 * half + 1) * 128];
        acc = __builtin_amdgcn_wmma_f32_16x16x4_f32(false, a, false, b,
                                                    (short)0, acc, false, false);
    }
    float* crow = C + tj * 16 + m;
#pragma unroll
    for (int r = 0; r < 8; ++r)
        crow[(ti * 16 + r + 8 * half) * 128] = acc[r];
}

// ---------------------------------------------------------------------------
// Stop ("no more edges") logits: out[E + g]
// ---------------------------------------------------------------------------
__global__ void stop_kernel(const float* __restrict__ graphC,
                            const float* __restrict__ stop_tail,
                            const float* __restrict__ W2s,
                            const float* __restrict__ b2s,
                            float* __restrict__ out, int E, int G) {
    const int lane = threadIdx.x & 31;
    const int wave = threadIdx.x >> 5;
    const int g = blockIdx.x * (blockDim.x >> 5) + wave;
    if (g >= G) return;
    const float* gc = graphC + (long)g * 128;
    float h0 = fmaxf(gc[lane] + stop_tail[lane], 0.0f);
    float h1 = fmaxf(gc[lane + 32] + stop_tail[lane + 32], 0.0f);
    float p = h0 * W2s[lane] + h1 * W2s[lane + 32];
    for (int off = 16; off > 0; off >>= 1) p += __shfl_xor(p, off, 32);
    if (lane == 0) out[E + g] = p + b2s[0];
}

// ---------------------------------------------------------------------------
// Edge kernel: one wave per edge. Gather graphC[g] + nodeC[t] (512B each,
// coalesced), add tail contributions, ReLU, two wave dot-products.
// ---------------------------------------------------------------------------
__global__ void edge_kernel(const int* __restrict__ targets,
                            const int* __restrict__ n2g,
                            const float* __restrict__ feats,   // [E,4]
                            const float* __restrict__ dist_tab,// [10]
                            const float* __restrict__ nodeC,   // [N,128]
                            const float* __restrict__ graphC,  // [G,128]
                            const float* __restrict__ bias,    // [128]
                            const float* __restrict__ Wtail,   // [4,128]
                            const float* __restrict__ W2s,     // [64]
                            const float* __restrict__ b2s,     // [1]
                            const float* __restrict__ W2t,     // [64,3]
                            const float* __restrict__ b2t,     // [3]
                            float* __restrict__ out, int E, int G) {
    __shared__ float sBias[128];
    __shared__ float sWt[4 * 128];
    __shared__ float sW2s[64];
    __shared__ float sW2t[64 * 3];
    for (int i = threadIdx.x; i < 128; i += blockDim.x) sBias[i] = bias[i];
    for (int i = threadIdx.x; i < 512; i += blockDim.x) sWt[i] = Wtail[i];
    for (int i = threadIdx.x; i < 64; i += blockDim.x) sW2s[i] = W2s[i];
    for (int i = threadIdx.x; i < 192; i += blockDim.x) sW2t[i] = W2t[i];
    __syncthreads();

    const int lane = threadIdx.x & 31;
    const int wave = threadIdx.x >> 5;
    const int e = blockIdx.x * (blockDim.x >> 5) + wave;
    if (e >= E) return;

    const int t = targets[e];
    const int g = n2g[t];
    const float f0 = feats[e * 4 + 0];
    const float f1 = feats[e * 4 + 1];
    const float f2 = feats[e * 4 + 2];
    const float f3 = feats[e * 4 + 3];
    const int di = (int)fminf(f0, 9.0f);
    const float dv = dist_tab[di];

    const float* gc = graphC + (long)g * 128;
    const float* nc = nodeC + (long)t * 128;

    float h[4];
#pragma unroll
    for (int q = 0; q < 4; ++q) {
        const int k = lane + 32 * q;
        float v = gc[k] + nc[k] + sBias[k]
                + dv * sWt[k] + f1 * sWt[128 + k]
                + f2 * sWt[256 + k] + f3 * sWt[384 + k];
        h[q] = fmaxf(v, 0.0f);
    }

    // scorer: hidden units 0..63 live in h[0],h[1]
    float ps = h[0] * sW2s[lane] + h[1] * sW2s[lane + 32];
    // type: hidden units 0..63 live in h[2],h[3]
    float p0 = h[2] * sW2t[lane * 3 + 0] + h[3] * sW2t[(lane + 32) * 3 + 0];
    float p1 = h[2] * sW2t[lane * 3 + 1] + h[3] * sW2t[(lane + 32) * 3 + 1];
    float p2 = h[2] * sW2t[lane * 3 + 2] + h[3] * sW2t[(lane + 32) * 3 + 2];
    for (int off = 16; off > 0; off >>= 1) {
        ps += __shfl_xor(ps, off, 32);
        p0 += __shfl_xor(p0, off, 32);
        p1 += __shfl_xor(p1, off, 32);
        p2 += __shfl_xor(p2, off, 32);
    }
    if (lane == 0) {
        out[e] = ps + b2s[0];
        float* tl = out + (long)(E + G) + (long)e * 3;
        tl[0] = p0 + b2t[0];
        tl[1] = p1 + b2t[1];
        tl[2] = p2 + b2t[2];
    }
}

// ---------------------------------------------------------------------------
extern "C" void kernel_launch(void* const* d_in, const int* in_sizes, int n_in,
                              void* d_out, int out_size, void* d_ws, size_t ws_size,
                              hipStream_t stream) {
    const float* mol    = (const float*)d_in[0];   // [G,128]
    const float* par    = (const float*)d_in[1];   // [G,128]
    const float* nodes  = (const float*)d_in[2];   // [N,128]
    const int*   focus  = (const int*)d_in[3];     // [G]
    const int*   n2g    = (const int*)d_in[4];     // [N]
    const int*   etgt   = (const int*)d_in[5];     // [E]
    const float* efeat  = (const float*)d_in[6];   // [E,4]
    const float* dtab   = (const float*)d_in[8];   // [10,1]
    const float* nomore = (const float*)d_in[9];   // [1,132]
    const float* sW1    = (const float*)d_in[10];  // [516,64]
    const float* sb1    = (const float*)d_in[11];  // [64]
    const float* sW2    = (const float*)d_in[12];  // [64,1]
    const float* sb2    = (const float*)d_in[13];  // [1]
    const float* tW1    = (const float*)d_in[14];  // [516,64]
    const float* tb1    = (const float*)d_in[15];  // [64]
    const float* tW2    = (const float*)d_in[16];  // [64,3]
    const float* tb2    = (const float*)d_in[17];  // [3]
    float* out = (float*)d_out;

    const int G = in_sizes[3];
    const int N = in_sizes[4];
    const int E = in_sizes[5];

    // workspace layout (floats)
    float* ws = (float*)d_ws;
    float* Wn        = ws;                       // 128*128   = 16384
    float* Wg        = Wn + 128 * 128;           // 384*128   = 49152
    float* Wtail     = Wg + 384 * 128;           // 4*128     = 512
    float* biasC     = Wtail + 4 * 128;          // 128
    float* stop_tail = biasC + 128;              // 64
    float* graphC    = stop_tail + 64 + 64;      // G*128 (64-float pad for align)
    float* nodeC     = graphC + (size_t)G * 128; // N*128

    // 1) pack weights
    {
        const int total = 128 * 128 + 384 * 128 + 4 * 128 + 128 + 64;
        pack_weights_kernel<<<(total + 255) / 256, 256, 0, stream>>>(
            sW1, sb1, tW1, tb1, nomore, Wn, Wg, Wtail, biasC, stop_tail);
    }
    // 2) nodeC = nodes @ Wn   (WMMA fp32)
    gemm_node_kernel<<<N / 16, 256, 0, stream>>>(nodes, Wn, nodeC);
    // 3) graphC = [mol|par|nodes[focus]] @ Wg   (WMMA fp32)
    gemm_graph_kernel<<<(G / 16) * 8 / 8, 256, 0, stream>>>(mol, par, nodes, focus, Wg, graphC);
    // 4) stop logits
    stop_kernel<<<(G + 7) / 8, 256, 0, stream>>>(graphC, stop_tail, sW2, sb2, out, E, G);
    // 5) edge logits + type logits
    edge_kernel<<<(E + 7) / 8, 256, 0, stream>>>(etgt, n2g, efeat, dtab, nodeC, graphC,
                                                 biasC, Wtail, sW2, sb2, tW2, tb2,
                                                 out, E, G);
}